// FreeConv2D_74921409511758
// MI455X (gfx1250) — compile-verified
//
#include <hip/hip_runtime.h>
#include <hip/hip_bf16.h>

// Problem constants (from reference)
#define BB 64
#define HH 64
#define WW 64
#define CC 64
#define UU 64
#define KK 5
#define SS 2
#define OH 30
#define OW 30

typedef __attribute__((ext_vector_type(2))) float v2f;
typedef __attribute__((ext_vector_type(8))) float v8f;

#define LDS_STRIDE 68  // 64 data dwords + 4 pad dwords -> 272B rows (16B aligned, bank-spread)
#define TILE_FLOATS (64 * LDS_STRIDE)
#define SMEM_BYTES (4 * TILE_FLOATS * 4)  // sX[2] + sW[2] = 69,632 B (dynamic LDS)

// CDNA5 async copy: one 16B chunk per lane, global -> LDS, tracked by ASYNCcnt.
// GV mode: INST_OFFSET is added to BOTH the LDS and the global address.
#define ASYNC16(lds, ga, OFF)                                              \
    asm volatile("global_load_async_to_lds_b128 %0, %1, off offset:" #OFF  \
                 :: "v"(lds), "v"(ga) : "memory")
#define ASYNC16_NT(lds, ga, OFF)                                           \
    asm volatile("global_load_async_to_lds_b128 %0, %1, off offset:" #OFF  \
                 " th:TH_LOAD_NT"                                          \
                 :: "v"(lds), "v"(ga) : "memory")
#define WAIT_ASYNC0() asm volatile("s_wait_asynccnt 0x0" ::: "memory")

__device__ __forceinline__ unsigned lds_addr32(const void* p) {
    // Flat shared-aperture address: low 32 bits are the wave-relative LDS byte offset.
    return (unsigned)(unsigned long long)(uintptr_t)p;
}

__global__ __launch_bounds__(256)
void FreeConv2D_kernel(const float* __restrict__ x,
                       const float* __restrict__ w,
                       const float* __restrict__ bias,
                       float* __restrict__ out) {
    // One workgroup per output spatial position: GEMM M=64(batch) x N=64(U) x K=1600
    const int hw = blockIdx.x;
    const int oh = hw / OW;
    const int ow = hw % OW;

    extern __shared__ float smem[];
    // Layout: [buf0 X][buf1 X][buf0 W][buf1 W], each TILE_FLOATS floats.

    const int tid  = threadIdx.x;   // 0..255 (8 waves of 32)
    const int wv   = tid >> 5;      // wave id 0..7
    const int lane = tid & 31;

    // Tile assignment: wave -> (M-tile mi in 0..3, two N-tiles nb, nb+1)
    const int mi = wv & 3;          // M tile index (batch 16-row group)
    const int nb = (wv >> 2) * 2;   // first of two N tile indices (0 or 2)

    // Fragment addressing (ISA 16x16x4 f32 layout):
    // lane L holds row (L%16); VGPR v holds K = k0 + v + 2*(L/16)
    const int l16   = lane & 15;
    const int khalf = (lane >> 4) * 2;
    const int arow  = mi * 16 + l16;          // A-fragment row in sX
    const int brow0 = nb * 16 + l16;          // B-fragment row (tile nb) in sW
    const int brow1 = brow0 + 16;             // B-fragment row (tile nb+1)

    // Staging assignment: 4 threads per row, 16 floats (4 x 16B async chunks) each
    const int ldRow = tid >> 2;               // 0..63
    const int ldCol = (tid & 3) * 16;         // 0,16,32,48

    const long xBase = ((long)ldRow * HH) * WW * CC;
    const long wBase = (((long)(oh * OW + ow) * UU + ldRow) * KK) * KK * CC;

    v8f acc0 = {};
    v8f acc1 = {};

    // Issue the async staging of one tap into LDS buffer `buf` (runs in background).
    auto issue_tap = [&](int tap, int buf) {
        const int i = tap / KK;
        const int j = tap % KK;
        const float* gx = x + xBase + ((long)(oh * SS + i) * WW + (ow * SS + j)) * CC + ldCol;
        const float* gw = w + wBase + ((long)i * KK + j) * CC + ldCol;
        const float* sXb = smem + buf * TILE_FLOATS;
        const float* sWb = smem + (2 + buf) * TILE_FLOATS;
        const unsigned lx = lds_addr32(sXb + ldRow * LDS_STRIDE + ldCol);
        const unsigned lw = lds_addr32(sWb + ldRow * LDS_STRIDE + ldCol);
        const unsigned long long ax = (unsigned long long)(uintptr_t)gx;
        const unsigned long long aw = (unsigned long long)(uintptr_t)gw;
        ASYNC16(lx, ax, 0);  ASYNC16(lx, ax, 16);
        ASYNC16(lx, ax, 32); ASYNC16(lx, ax, 48);
        // Weights are single-use (369 MB stream): non-temporal so x stays L2-resident.
        ASYNC16_NT(lw, aw, 0);  ASYNC16_NT(lw, aw, 16);
        ASYNC16_NT(lw, aw, 32); ASYNC16_NT(lw, aw, 48);
    };

    issue_tap(0, 0);

    #pragma unroll 1
    for (int tap = 0; tap < KK * KK; ++tap) {
        const int cur = tap & 1;
        // Own async copies done -> barrier makes the whole buffer visible to all waves,
        // and guarantees everyone finished reading buf cur^1 (tap-1's compute).
        WAIT_ASYNC0();
        __syncthreads();

        // Overlap next tap's staging with this tap's WMMAs.
        if (tap + 1 < KK * KK) issue_tap(tap + 1, cur ^ 1);

        const float* bX = smem + cur * TILE_FLOATS;
        const float* bW = smem + (2 + cur) * TILE_FLOATS;

        // ---- reduce over c = 0..63 in K-steps of 4 with f32 WMMA ----
        #pragma unroll
        for (int k0 = 0; k0 < CC; k0 += 4) {
            const int kk = k0 + khalf;
            const v2f a  = *(const v2f*)(bX + arow  * LDS_STRIDE + kk);
            const v2f b0 = *(const v2f*)(bW + brow0 * LDS_STRIDE + kk);
            const v2f b1 = *(const v2f*)(bW + brow1 * LDS_STRIDE + kk);
            acc0 = __builtin_amdgcn_wmma_f32_16x16x4_f32(
                false, a, false, b0, (short)0, acc0, false, false);
            acc1 = __builtin_amdgcn_wmma_f32_16x16x4_f32(
                false, a, false, b1, (short)0, acc1, false, false);
        }
    }

    // ---- epilogue: C layout -> out[b, oh, ow, u] (+bias) ----
    const int u0 = nb * 16 + l16;       // N for acc0
    const int u1 = u0 + 16;             // N for acc1
    const float bias0 = bias[(oh * OW + ow) * UU + u0];
    const float bias1 = bias[(oh * OW + ow) * UU + u1];
    const int mhi = 8 * (lane >> 4);    // lanes 16..31 hold M = r+8

    #pragma unroll
    for (int r = 0; r < 8; ++r) {
        const int b = mi * 16 + r + mhi;
        const long o = (((long)b * OH + oh) * OW + ow) * UU;
        out[o + u0] = acc0[r] + bias0;
        out[o + u1] = acc1[r] + bias1;
    }
}

extern "C" void kernel_launch(void* const* d_in, const int* in_sizes, int n_in,
                              void* d_out, int out_size, void* d_ws, size_t ws_size,
                              hipStream_t stream) {
    const float* x    = (const float*)d_in[0];
    const float* w    = (const float*)d_in[1];
    const float* bias = (const float*)d_in[2];
    float* out        = (float*)d_out;
    (void)in_sizes; (void)n_in; (void)out_size; (void)d_ws; (void)ws_size;

    dim3 grid(OH * OW);   // 900 workgroups, one per output spatial position
    dim3 block(256);      // 8 wave32s
    FreeConv2D_kernel<<<grid, block, SMEM_BYTES, stream>>>(x, w, bias, out);
}